// BatchBlur_SV_NAF_54382875902266
// MI455X (gfx1250) — compile-verified
//
#include <hip/hip_runtime.h>
#include <stdint.h>

#define NPIX   65536        // 256*256
#define KK     361          // 19*19
#define KFN    (KK * NPIX)  // 23,658,496 elements
#define PADW   274
#define PADN   (PADW * PADW)

#if defined(__HIP_DEVICE_COMPILE__) && \
    __has_builtin(__builtin_amdgcn_global_load_async_to_lds_b32) && \
    __has_builtin(__builtin_amdgcn_s_wait_asynccnt)
#define USE_ASYNC_LDS 1
#else
#define USE_ASYNC_LDS 0
#endif

#if defined(__HIP_DEVICE_COMPILE__) && \
    __has_builtin(__builtin_amdgcn_wmma_f32_16x16x4_f32)
#define USE_WMMA_S1 1
#else
#define USE_WMMA_S1 0
#endif

typedef __attribute__((ext_vector_type(2))) float v2f;
typedef __attribute__((ext_vector_type(8))) float v8f;

__device__ __forceinline__ float wave_sum32(float v) {
#pragma unroll
  for (int off = 16; off > 0; off >>= 1) v += __shfl_xor(v, off, 32);
  return v;
}

// Reflect-pad channel 2 of input into pad (274x274).
__global__ void k_pad(const float* __restrict__ inp, float* __restrict__ pad) {
  int i = blockIdx.x * 256 + threadIdx.x;
  if (i >= PADN) return;
  int py = i / PADW, px = i - py * PADW;
  int sy = py - 9; sy = sy < 0 ? -sy : sy; sy = sy > 255 ? 510 - sy : sy;
  int sx = px - 9; sx = sx < 0 ? -sx : sx; sx = sx > 255 ? 510 - sx : sx;
  pad[i] = inp[2 * NPIX + sy * 256 + sx];
}

// S1R[q] = 1 / sum_{T<361} kf[T*65536 + q].
// Column-sum of the (361,65536) matrix == ones-vector matvec -> WMMA path:
// A (16x4) = ones, B (4x16) = kf tile, accumulate D over 90 K=4 steps.
// Launch: 512 blocks x 256 threads; each wave owns 16 consecutive columns.
__global__ void k_s1r(const float* __restrict__ kf, float* __restrict__ s1r) {
  int lane = threadIdx.x & 31;
  int wv   = threadIdx.x >> 5;
  int q0   = (blockIdx.x * 8 + wv) * 16;   // 16 columns per wave
#if USE_WMMA_S1
  int n     = lane & 15;                   // column within group
  int khalf = lane >> 4;                   // lanes 16-31 hold K=2,3
  const float* base = kf + q0 + n;
  v8f acc = {};
  v2f a; a.x = 1.f; a.y = 1.f;             // ones A-matrix
  for (int T = 0; T < 360; T += 4) {
    int k0 = T + (khalf << 1);
    v2f b;
    b.x = base[(size_t)k0 * NPIX];
    b.y = base[(size_t)(k0 + 1) * NPIX];
    acc = __builtin_amdgcn_wmma_f32_16x16x4_f32(
        /*neg_a=*/false, a, /*neg_b=*/false, b,
        /*c_mod=*/(short)0, acc, /*reuse_a=*/false, /*reuse_b=*/false);
  }
  if (lane < 16) {                          // D row 0: acc[0], lanes 0..15
    float s = acc[0] + base[(size_t)360 * NPIX];  // leftover plane T=360
    s1r[q0 + lane] = 1.0f / s;
  }
#else
  if (lane < 16) {
    float a = 0.f;
#pragma unroll 8
    for (int T = 0; T < KK; ++T) a += kf[T * NPIX + q0 + lane];
    s1r[q0 + lane] = 1.0f / a;
  }
#endif
}

// Tiled transpose of the (65536,361) row-major view of kf into Wt[t*65536+r],
// scaled by S1R on the coalesced (read) side; also emits deterministic per-row
// partial sums for S2. Reads go global->LDS via the CDNA5 async path.
__global__ void k_transpose(const float* __restrict__ kf,
                            const float* __restrict__ s1r,
                            float* __restrict__ wt,
                            float* __restrict__ part) {
  __shared__ float tile[32][33];
  int tid  = threadIdx.x;
  int lane = tid & 31, wv = tid >> 5;
  int R = blockIdx.x << 5;  // r-tile base (0..65504)
  int C = blockIdx.y << 5;  // t-tile base (0..352)

  // phase 1: raw tile load V[r][t] = kf[r*361 + t], rows striped over waves
#pragma unroll
  for (int k = 0; k < 4; ++k) {
    int row = wv + (k << 3);
    int c = C + lane;
    if (c < KK) {
      int idx = (R + row) * KK + c;
#if USE_ASYNC_LDS
      __builtin_amdgcn_global_load_async_to_lds_b32(
          (__attribute__((address_space(1))) int*)(uintptr_t)(kf + idx),
          (__attribute__((address_space(3))) int*)(uint32_t)(uintptr_t)&tile[row][lane],
          0, 0);
#else
      tile[row][lane] = kf[idx];
#endif
    } else {
      tile[row][lane] = 0.f;
    }
  }
#if USE_ASYNC_LDS
  __builtin_amdgcn_s_wait_asynccnt(0);
#endif
  __syncthreads();

  // phase 1.5: scale by S1R (index u & 0xFFFF is coalesced along t here),
  // and produce per-row partial sums for S2 (wave owns each row -> shuffle).
#pragma unroll
  for (int k = 0; k < 4; ++k) {
    int row = wv + (k << 3);
    int c = C + lane;
    int u = (R + row) * KK + c;
    float v = tile[row][lane];
    v = (c < KK) ? v * s1r[u & (NPIX - 1)] : 0.f;
    tile[row][lane] = v;
    float s = wave_sum32(v);
    if (lane == 0) part[(R + row) * 12 + blockIdx.y] = s;
  }
  __syncthreads();

  // phase 2: coalesced transposed write: Wt[t*65536 + r] = tile[r-R][t-C]
#pragma unroll
  for (int k = 0; k < 4; ++k) {
    int tt = wv + (k << 3);
    int t = C + tt;
    if (t < KK) wt[t * NPIX + (R + lane)] = tile[lane][tt];
  }
}

// S2R[q] = 1 / sum of 12 tile partials (deterministic reduction)
__global__ void k_s2fin(const float* __restrict__ part, float* __restrict__ s2r) {
  int q = blockIdx.x * 256 + threadIdx.x;
  float s = 0.f;
#pragma unroll
  for (int cb = 0; cb < 12; ++cb) s += part[q * 12 + cb];
  s2r[q] = 1.0f / s;
}

// Final: wave per pixel p. w = Wt[m]*S2R[m&0xFFFF]; out = (Σ dg*w)/(Σ w).
__global__ void k_final(const float* __restrict__ wt,
                        const float* __restrict__ s2r,
                        const float* __restrict__ pad,
                        float* __restrict__ out) {
  int tid = threadIdx.x;
  int lane = tid & 31;
  int p = (blockIdx.x << 3) + (tid >> 5);
  float sw = 0.f, so = 0.f;
  for (int j = lane; j < KK; j += 32) {
    int m = p * KK + j;
    int t = m >> 16;
    int r = m & (NPIX - 1);
    float w = wt[m] * s2r[r];
    sw += w;
    int y = r >> 8, x = r & 255;
    int ta = t / 19, tb = t - ta * 19;
    so += w * pad[(y + ta) * PADW + (x + tb)];
  }
  sw = wave_sum32(sw);
  so = wave_sum32(so);
  if (lane == 0) out[p] = so / sw;
}

// ---- fallback path (small workspace): direct gather, no Wt materialization
__global__ void k_s2g(const float* __restrict__ kf, const float* __restrict__ s1r,
                      float* __restrict__ s2r) {
  int lane = threadIdx.x & 31;
  int q = (blockIdx.x << 3) + (threadIdx.x >> 5);
  float s = 0.f;
  for (int j = lane; j < KK; j += 32) {
    int i = q * KK + j;
    s += kf[i] * s1r[i & (NPIX - 1)];
  }
  s = wave_sum32(s);
  if (lane == 0) s2r[q] = 1.0f / s;
}

__global__ void k_final_g(const float* __restrict__ kf, const float* __restrict__ s1r,
                          const float* __restrict__ s2r, const float* __restrict__ pad,
                          float* __restrict__ out) {
  int lane = threadIdx.x & 31;
  int p = (blockIdx.x << 3) + (threadIdx.x >> 5);
  float sw = 0.f, so = 0.f;
  for (int j = lane; j < KK; j += 32) {
    int m = p * KK + j;
    int t = m >> 16;
    int r = m & (NPIX - 1);
    int u = r * KK + t;
    float w = kf[u] * s1r[u & (NPIX - 1)] * s2r[r];
    sw += w;
    int y = r >> 8, x = r & 255;
    int ta = t / 19, tb = t - ta * 19;
    so += w * pad[(y + ta) * PADW + (x + tb)];
  }
  sw = wave_sum32(sw);
  so = wave_sum32(so);
  if (lane == 0) out[p] = so / sw;
}

extern "C" void kernel_launch(void* const* d_in, const int* in_sizes, int n_in,
                              void* d_out, int out_size, void* d_ws, size_t ws_size,
                              hipStream_t stream) {
  (void)in_sizes; (void)n_in; (void)out_size;
  const float* inp = (const float*)d_in[0];   // (1,3,256,256) f32
  const float* kf  = (const float*)d_in[1];   // (19,19,256,256) f32
  float* out = (float*)d_out;                 // (1,1,256,256) f32

  char* ws = (char*)d_ws;
  float* S1R  = (float*)(ws + 0);                     // 256 KB
  float* S2R  = (float*)(ws + (256u << 10));          // 256 KB
  float* PAD  = (float*)(ws + (512u << 10));          // ~300 KB
  float* PART = (float*)(ws + (832u << 10));          // 3 MB  (65536*12 f32)
  float* WT   = (float*)(ws + (4u << 20));            // 94.6 MB

  bool fast = ws_size >= (size_t)(4u << 20) + (size_t)KFN * 4u;

  k_pad<<<(PADN + 255) / 256, 256, 0, stream>>>(inp, PAD);
  k_s1r<<<512, 256, 0, stream>>>(kf, S1R);   // 512*8 waves * 16 cols = 65536

  if (fast) {
    k_transpose<<<dim3(NPIX / 32, 12), 256, 0, stream>>>(kf, S1R, WT, PART);
    k_s2fin<<<NPIX / 256, 256, 0, stream>>>(PART, S2R);
    k_final<<<NPIX / 8, 256, 0, stream>>>(WT, S2R, PAD, out);
  } else {
    k_s2g<<<NPIX / 8, 256, 0, stream>>>(kf, S1R, S2R);
    k_final_g<<<NPIX / 8, 256, 0, stream>>>(kf, S1R, S2R, PAD, out);
  }
}